// SPL_86131274154226
// MI455X (gfx1250) — compile-verified
//
#include <hip/hip_runtime.h>
#include <hip/hip_bf16.h>
#include <stdint.h>

// MI455X (gfx1250, wave32) kernel:
//   Pass 1: per-row mean((out-y)^2), one block per row, streaming b128 NT loads,
//           wave reduction via V_WMMA_F32_16X16X4_F32 (B = ones -> row sums).
//   Pass 2: exact top-k(4096) sum via bitwise radix-select on monotonic uint keys.
// Bandwidth bound: 512 MB @ 23.3 TB/s ~= 22 us floor; everything else is noise.

typedef float f32x2 __attribute__((ext_vector_type(2)));
typedef float f32x4 __attribute__((ext_vector_type(4)));
typedef float f32x8 __attribute__((ext_vector_type(8)));

#define N_ROWS 8192
#define N_COLS 8192
#define K_TOP  4096
#define BLK    256

// ---------------------------------------------------------------------------
// Pass 1: per-row sum of squared differences.
// ---------------------------------------------------------------------------
__global__ __launch_bounds__(BLK) void rowmse_kernel(const float* __restrict__ outp,
                                                     const float* __restrict__ yp,
                                                     float* __restrict__ per_sample) {
  const int row = blockIdx.x;
  const int tid = threadIdx.x;

  const f32x4* o4 = (const f32x4*)(outp + (size_t)row * N_COLS);
  const f32x4* y4 = (const f32x4*)(yp   + (size_t)row * N_COLS);

  // 8192 floats/row = 2048 float4; 256 threads * 8 iters, stride-256 coalesced.
  float acc0 = 0.0f, acc1 = 0.0f;
#pragma unroll
  for (int i = 0; i < 8; ++i) {
    const int idx = tid + i * BLK;
    f32x4 a = __builtin_nontemporal_load(&o4[idx]);   // stream: no reuse, NT hint
    f32x4 b = __builtin_nontemporal_load(&y4[idx]);
    f32x4 d = a - b;
    float s = d.x * d.x + d.y * d.y + d.z * d.z + d.w * d.w;
    if (i < 4) acc0 += s; else acc1 += s;             // two accumulators for ILP
  }

  // Wave32 reduction using the f32 WMMA:
  //   A(16x4): lane m -> {K0=acc0, K1=acc1}, lane m+16 -> {K2=acc0, K3=acc1}
  //   B(4x16) = all ones  =>  D[m][n] = acc0[m]+acc1[m]+acc0[m+16]+acc1[m+16]
  // Full f32 precision, EXEC is all-ones here (no divergence).
  float wave_total;
#if __has_builtin(__builtin_amdgcn_wmma_f32_16x16x4_f32)
  {
    f32x2 a; a.x = acc0; a.y = acc1;
    f32x2 b; b.x = 1.0f; b.y = 1.0f;
    f32x8 c = {};
    f32x8 d = __builtin_amdgcn_wmma_f32_16x16x4_f32(
        /*neg_a=*/false, a, /*neg_b=*/false, b,
        /*c_mod=*/(short)0, c, /*reuse_a=*/false, /*reuse_b=*/false);
    // Lane L, VGPR v holds S[v + (L<16 ? 0 : 8)] (broadcast over N).
    float half_sum = d[0] + d[1] + d[2] + d[3] + d[4] + d[5] + d[6] + d[7];
    wave_total = half_sum + __shfl_xor(half_sum, 16, 32);  // S[0..7] + S[8..15]
  }
#else
  {
    float v = acc0 + acc1;
#pragma unroll
    for (int off = 16; off >= 1; off >>= 1) v += __shfl_xor(v, off, 32);
    wave_total = v;
  }
#endif

  __shared__ float s_wave[BLK / 32];
  const int lane = tid & 31, wid = tid >> 5;
  if (lane == 0) s_wave[wid] = wave_total;
  __syncthreads();
  if (tid == 0) {
    float t = 0.0f;
#pragma unroll
    for (int i = 0; i < BLK / 32; ++i) t += s_wave[i];
    per_sample[row] = t * (1.0f / (float)N_COLS);
  }
}

// ---------------------------------------------------------------------------
// Pass 2: sum of the 4096 largest of 8192 values. Single block, LDS resident.
// ---------------------------------------------------------------------------
__device__ __forceinline__ uint32_t key_of(float f) {
  uint32_t u = __float_as_uint(f);
  return (u & 0x80000000u) ? ~u : (u | 0x80000000u);  // monotonic order map
}
__device__ __forceinline__ float val_of(uint32_t u) {
  return (u & 0x80000000u) ? __uint_as_float(u ^ 0x80000000u) : __uint_as_float(~u);
}

__global__ __launch_bounds__(BLK) void topk_sum_kernel(const float* __restrict__ per_sample,
                                                       float* __restrict__ result) {
  __shared__ uint32_t s_key[N_ROWS];          // 32 KB of the 320 KB WGP LDS
  __shared__ unsigned int s_cnt;
  __shared__ float        s_red[BLK];
  __shared__ unsigned int s_cred[BLK];

  const int tid = threadIdx.x;

  for (int i = tid; i < N_ROWS; i += BLK) s_key[i] = key_of(per_sample[i]);
  __syncthreads();

  // Bitwise select: T = max{x : count(key >= x) >= K_TOP} == k-th largest key.
  uint32_t T = 0;
  for (int bit = 31; bit >= 0; --bit) {
    const uint32_t cand = T | (1u << bit);
    if (tid == 0) s_cnt = 0;
    __syncthreads();
    unsigned int c = 0;
    for (int i = tid; i < N_ROWS; i += BLK) c += (s_key[i] >= cand) ? 1u : 0u;
    atomicAdd(&s_cnt, c);
    __syncthreads();
    if (s_cnt >= K_TOP) T = cand;               // same value in every thread
    __syncthreads();
  }

  // Exact top-k sum with tie handling: sum(v > T) + (k - count_gt) * T.
  float sum_gt = 0.0f;
  unsigned int cnt_gt = 0;
  for (int i = tid; i < N_ROWS; i += BLK) {
    const uint32_t u = s_key[i];
    if (u > T) { sum_gt += val_of(u); ++cnt_gt; }
  }
  s_red[tid]  = sum_gt;
  s_cred[tid] = cnt_gt;
  __syncthreads();
#pragma unroll
  for (int off = BLK / 2; off >= 1; off >>= 1) {  // deterministic tree reduce
    if (tid < off) {
      s_red[tid]  += s_red[tid + off];
      s_cred[tid] += s_cred[tid + off];
    }
    __syncthreads();
  }
  if (tid == 0) {
    result[0] = s_red[0] + (float)(K_TOP - s_cred[0]) * val_of(T);
  }
}

// ---------------------------------------------------------------------------
extern "C" void kernel_launch(void* const* d_in, const int* in_sizes, int n_in,
                              void* d_out, int out_size, void* d_ws, size_t ws_size,
                              hipStream_t stream) {
  const float* outp = (const float*)d_in[0];
  const float* yp   = (const float*)d_in[1];
  float* per_sample = (float*)d_ws;            // 8192 floats of scratch

  rowmse_kernel<<<N_ROWS, BLK, 0, stream>>>(outp, yp, per_sample);
  topk_sum_kernel<<<1, BLK, 0, stream>>>(per_sample, (float*)d_out);
}